// SpectralGCN_50294067036722
// MI455X (gfx1250) — compile-verified
//
#include <hip/hip_runtime.h>

// ---------------------------------------------------------------------------
// SpectralGCN for MI455X (gfx1250, wave32).
// fp32 end-to-end using V_WMMA_F32_16X16X4_F32 for all GEMM-shaped stages.
// SVD replaced by subspace iteration on G = Lap*Lap^T (same U,S as SVD of Lap);
// rank-K spectral conv applied in factored form Us*(Uo^T*(h@w)) (no dense M).
// X staging for the power-iteration GEMMs uses the Tensor Data Mover
// (tensor_load_to_lds + s_wait_tensorcnt) when the builtin is available.
// Workspace use: ~26 MB.
// ---------------------------------------------------------------------------

typedef float v2f __attribute__((ext_vector_type(2)));
typedef float v8f __attribute__((ext_vector_type(8)));
typedef unsigned int v4u __attribute__((ext_vector_type(4)));
typedef int v8i __attribute__((ext_vector_type(8)));
typedef int v4i __attribute__((ext_vector_type(4)));

#define BSZ   64
#define NN    1024
#define FE    8
#define TT    12
#define HH    32
#define BN    65536   // BSZ*NN
#define KRANK 10
#define NEIG  16      // padded subspace width (one WMMA tile)

__device__ __forceinline__ v8f wmma4(v2f a, v2f b, v8f c) {
  // D = A(16x4,f32) * B(4x16,f32) + C(16x16,f32)
  return __builtin_amdgcn_wmma_f32_16x16x4_f32(false, a, false, b, (short)0, c,
                                               false, false);
}

// ===========================================================================
// 1) Temporal GRU: one thread per (b,n) row, weights broadcast from LDS.
// ===========================================================================
__global__ void gru_kernel(const float* __restrict__ x,
                           const float* __restrict__ wih,
                           const float* __restrict__ whh,
                           const float* __restrict__ bih,
                           const float* __restrict__ bhh,
                           float* __restrict__ Hout) {
  __shared__ float sWih[3 * HH * FE];   // 768
  __shared__ float sWhh[3 * HH * HH];   // 3072
  __shared__ float sBih[3 * HH];
  __shared__ float sBhh[3 * HH];
  const int tid = threadIdx.x;
  for (int i = tid; i < 3 * HH * FE; i += 128) sWih[i] = wih[i];
  for (int i = tid; i < 3 * HH * HH; i += 128) sWhh[i] = whh[i];
  if (tid < 3 * HH) { sBih[tid] = bih[tid]; sBhh[tid] = bhh[tid]; }
  __syncthreads();

  const int row = blockIdx.x * 128 + tid;          // < BN, grid exact
  const float* xp = x + (size_t)row * (FE * TT);   // x[b,n,f,t]
  float h[HH], r[HH], nh[HH], xv[FE];
#pragma unroll
  for (int k = 0; k < HH; ++k) h[k] = 0.0f;

  for (int t = 0; t < TT; ++t) {
#pragma unroll
    for (int f = 0; f < FE; ++f) xv[f] = xp[f * TT + t];
    // r gate (needs old h for all lanes of n gate)
    for (int j = 0; j < HH; ++j) {
      float g = sBih[j] + sBhh[j];
#pragma unroll
      for (int f = 0; f < FE; ++f) g += xv[f] * sWih[j * FE + f];
      for (int k = 0; k < HH; ++k) g += h[k] * sWhh[j * HH + k];
      r[j] = 1.0f / (1.0f + __expf(-g));
    }
    // z, n, update (n keeps x-part and h-part separate: tanh(xn + r*hn))
    for (int j = 0; j < HH; ++j) {
      float gz = sBih[HH + j] + sBhh[HH + j];
      float gxn = sBih[2 * HH + j];
      float ghn = sBhh[2 * HH + j];
#pragma unroll
      for (int f = 0; f < FE; ++f) {
        gz  += xv[f] * sWih[(HH + j) * FE + f];
        gxn += xv[f] * sWih[(2 * HH + j) * FE + f];
      }
      for (int k = 0; k < HH; ++k) {
        gz  += h[k] * sWhh[(HH + j) * HH + k];
        ghn += h[k] * sWhh[(2 * HH + j) * HH + k];
      }
      float z = 1.0f / (1.0f + __expf(-gz));
      float n = tanhf(gxn + r[j] * ghn);
      nh[j] = (1.0f - z) * n + z * h[j];
    }
    for (int j = 0; j < HH; ++j) h[j] = nh[j];
  }
  float* op = Hout + (size_t)row * HH;
  for (int j = 0; j < HH; ++j) op[j] = h[j];
}

// ===========================================================================
// 2) Graph build: A scatter, degrees, normalized Laplacian (+ its transpose).
// ===========================================================================
__global__ void zero_kernel(float* __restrict__ p, int n) {
  int i = blockIdx.x * 256 + threadIdx.x;
  if (i < n) p[i] = 0.0f;
}
__global__ void scatter_kernel(const int* __restrict__ ei,
                               const float* __restrict__ ew,
                               float* __restrict__ A, int E) {
  int e = blockIdx.x * 256 + threadIdx.x;
  if (e < E) atomicAdd(&A[(size_t)ei[e] * NN + ei[E + e]], ew[e]);
}
__global__ void degree_kernel(const float* __restrict__ A,
                              float* __restrict__ dis) {
  int n = blockIdx.x * 256 + threadIdx.x;
  if (n < NN) {
    float s = 0.0f;
    for (int m = 0; m < NN; ++m) s += A[(size_t)n * NN + m];
    dis[n] = rsqrtf(s);   // weights >= 0.1 with self-loops -> s > 0
  }
}
__global__ void lap_kernel(const float* __restrict__ A,
                           const float* __restrict__ dis,
                           float* __restrict__ L, float* __restrict__ LT) {
  int idx = blockIdx.x * 256 + threadIdx.x;   // NN*NN exact
  int n = idx >> 10, m = idx & (NN - 1);
  float v = ((n == m) ? 1.0f : 0.0f) - dis[n] * A[idx] * dis[m];
  L[idx] = v;
  LT[(size_t)m * NN + n] = v;
}

// ===========================================================================
// 3) Subspace iteration machinery.
// ===========================================================================
__global__ void initx_kernel(float* __restrict__ X) {
  int i = blockIdx.x * 256 + threadIdx.x;   // NN*NEIG exact
  unsigned u = (unsigned)i * 2654435761u;
  u ^= u >> 16; u *= 2246822519u; u ^= u >> 13;
  X[i] = ((u & 0xFFFFu) * (1.0f / 32768.0f)) - 1.0f;
}

// Out[1024x16] = Amat[1024x1024] @ Bx[1024x16], fp32 WMMA, X staged in LDS
// by the Tensor Data Mover (64 KB contiguous tile) when available.
// grid 16 x 128 threads: 64 waves == 64 row tiles, all waves fully active.
__global__ void gemm_nk16_kernel(const float* __restrict__ Amat,
                                 const float* __restrict__ Bx,
                                 float* __restrict__ Out) {
  __shared__ float sX[NN * NEIG];   // 64 KB
  const int tid = threadIdx.x;

#if defined(__gfx1250__) && __has_builtin(__builtin_amdgcn_tensor_load_to_lds)
  if (tid < 32) {   // wave 0 issues the DMA once (TDM ignores EXEC)
    const unsigned lds = (unsigned)(size_t)(__attribute__((address_space(3))) void*)sX;
    const unsigned long long ga = (unsigned long long)(size_t)Bx;
    // D# group 0: count=1, lds_addr, global_addr, type=2 ("image")
    v4u g0;
    g0[0] = 1u;                                   // count=1, gather off
    g0[1] = lds;                                  // lds_addr (bytes)
    g0[2] = (unsigned)ga;                         // global_addr[31:0]
    g0[3] = (unsigned)((ga >> 32) & 0x01FFFFFFu) | 0x80000000u; // [56:32]|type=2
    // D# group 1: data_size=4B (code 2); 2D tile 16384x1 of the 16384-elem
    // tensor, stride 16384 (elements)
    v8i g1;
    g1[0] = 0x20000;      // workgroup_mask=0, data_size=2, no flags
    g1[1] = 0x40000000;   // tensor_dim0[15:0]=16384 in bits[63:48]
    g1[2] = 0x00010000;   // tensor_dim0[31:16]=0, tensor_dim1[15:0]=1
    g1[3] = 0x40000000;   // tensor_dim1[31:16]=0, tile_dim0=16384
    g1[4] = 0x00000001;   // tile_dim1=1, tile_dim2=0
    g1[5] = 16384;        // tensor_dim0_stride[31:0]
    g1[6] = 0;            // stride hi, tensor_dim1_stride lo
    g1[7] = 0;
    v4i g2 = {0, 0, 0, 0};
    v4i g3 = {0, 0, 0, 0};
    v8i g4 = {0, 0, 0, 0, 0, 0, 0, 0};   // extra group (6-arg clang-23 form)
    __builtin_amdgcn_tensor_load_to_lds(g0, g1, g2, g3, g4, 0);
    __builtin_amdgcn_s_wait_tensorcnt(0);
  }
  __syncthreads();
#else
  for (int i = tid; i < NN * NEIG; i += 128) sX[i] = Bx[i];
  __syncthreads();
#endif

  const int wave = tid >> 5, lane = tid & 31;
  const int rt = blockIdx.x * 4 + wave;
  const int half = lane >> 4, c = lane & 15;
  const float* arow = Amat + (size_t)(rt * 16 + c) * NN + 2 * half;
  v8f acc = {0, 0, 0, 0, 0, 0, 0, 0};
#pragma unroll 4
  for (int kc = 0; kc < NN / 4; ++kc) {
    const int k0 = kc * 4;
    v2f a = *(const v2f*)(arow + k0);           // A frag: row c, K = k0+2h..+1
    v2f b;
    b[0] = sX[(k0 + 2 * half) * NEIG + c];      // B frag: col c, K = k0+2h
    b[1] = sX[(k0 + 2 * half + 1) * NEIG + c];
    acc = wmma4(a, b, acc);
  }
#pragma unroll
  for (int rr = 0; rr < 8; ++rr)
    Out[(size_t)(rt * 16 + rr + 8 * half) * NEIG + c] = acc[rr];
}

// In-place modified Gram-Schmidt on X[1024x16]; single 256-thread block.
__global__ void qr_kernel(float* __restrict__ X) {
  __shared__ float red[256];
  const int tid = threadIdx.x;
  for (int j = 0; j < NEIG; ++j) {
    float p = 0.0f;
    for (int n = tid; n < NN; n += 256) { float v = X[n * NEIG + j]; p += v * v; }
    red[tid] = p; __syncthreads();
    for (int s = 128; s > 0; s >>= 1) { if (tid < s) red[tid] += red[tid + s]; __syncthreads(); }
    float inv = rsqrtf(red[0] + 1e-20f); __syncthreads();
    for (int n = tid; n < NN; n += 256) X[n * NEIG + j] *= inv;
    __syncthreads();
    for (int k = j + 1; k < NEIG; ++k) {
      p = 0.0f;
      for (int n = tid; n < NN; n += 256) p += X[n * NEIG + j] * X[n * NEIG + k];
      red[tid] = p; __syncthreads();
      for (int s = 128; s > 0; s >>= 1) { if (tid < s) red[tid] += red[tid + s]; __syncthreads(); }
      float d = red[0]; __syncthreads();
      for (int n = tid; n < NN; n += 256) X[n * NEIG + k] -= d * X[n * NEIG + j];
      __syncthreads();
    }
  }
}

// T16 = X^T Z  (Rayleigh-Ritz projection), one thread per entry.
__global__ void t16_kernel(const float* __restrict__ X,
                           const float* __restrict__ Z,
                           float* __restrict__ T16) {
  const int tid = threadIdx.x;          // 256 threads
  const int i = tid >> 4, j = tid & 15;
  float s = 0.0f;
  for (int n = 0; n < NN; ++n) s += X[n * NEIG + i] * Z[n * NEIG + j];
  T16[tid] = s;
}

// Cyclic Jacobi eigensolver on symmetric 16x16; sorted descending.
__global__ void jacobi16_kernel(const float* __restrict__ T,
                                float* __restrict__ Vs,
                                float* __restrict__ lam) {
  if (threadIdx.x != 0) return;
  float a[NEIG][NEIG], v[NEIG][NEIG];
  for (int i = 0; i < NEIG; ++i)
    for (int j = 0; j < NEIG; ++j) {
      a[i][j] = 0.5f * (T[i * NEIG + j] + T[j * NEIG + i]);
      v[i][j] = (i == j) ? 1.0f : 0.0f;
    }
  for (int sweep = 0; sweep < 12; ++sweep)
    for (int p = 0; p < NEIG - 1; ++p)
      for (int q = p + 1; q < NEIG; ++q) {
        float apq = a[p][q];
        if (fabsf(apq) < 1e-12f) continue;
        float tau = (a[q][q] - a[p][p]) / (2.0f * apq);
        float t = (tau >= 0.0f ? 1.0f : -1.0f) / (fabsf(tau) + sqrtf(1.0f + tau * tau));
        float c = rsqrtf(1.0f + t * t), s = t * c;
        for (int k = 0; k < NEIG; ++k) {
          float akp = a[k][p], akq = a[k][q];
          a[k][p] = c * akp - s * akq; a[k][q] = s * akp + c * akq;
        }
        for (int k = 0; k < NEIG; ++k) {
          float apk = a[p][k], aqk = a[q][k];
          a[p][k] = c * apk - s * aqk; a[q][k] = s * apk + c * aqk;
        }
        for (int k = 0; k < NEIG; ++k) {
          float vkp = v[k][p], vkq = v[k][q];
          v[k][p] = c * vkp - s * vkq; v[k][q] = s * vkp + c * vkq;
        }
      }
  float ev[NEIG]; int perm[NEIG];
  for (int i = 0; i < NEIG; ++i) { ev[i] = a[i][i]; perm[i] = i; }
  for (int i = 0; i < NEIG; ++i) {
    int best = i;
    for (int j = i + 1; j < NEIG; ++j) if (ev[perm[j]] > ev[perm[best]]) best = j;
    int tmp = perm[i]; perm[i] = perm[best]; perm[best] = tmp;
  }
  for (int j = 0; j < NEIG; ++j) {
    lam[j] = ev[perm[j]];
    for (int k = 0; k < NEIG; ++k) Vs[k * NEIG + j] = v[k][perm[j]];
  }
}

// Uo = X @ Vsorted ; Us = Uo * diag(sqrt(lam_j) for j<K else 0).
__global__ void ritz_kernel(const float* __restrict__ X,
                            const float* __restrict__ Vs,
                            const float* __restrict__ lam,
                            float* __restrict__ Uo, float* __restrict__ Us) {
  int idx = blockIdx.x * 256 + threadIdx.x;   // NN*NEIG exact
  int n = idx >> 4, j = idx & 15;
  float s = 0.0f;
#pragma unroll
  for (int k = 0; k < NEIG; ++k) s += X[n * NEIG + k] * Vs[k * NEIG + j];
  Uo[idx] = s;
  float sv = (j < KRANK) ? sqrtf(fmaxf(lam[j], 0.0f)) : 0.0f;
  Us[idx] = s * sv;
}

// ===========================================================================
// 4) Spectral conv layers (factored): HW = H@w ; T1 = Uo^T@HW' ; H += relu(Us@T1)
// ===========================================================================
// HW[bn,k] = sum_h H[bn,h] * w[h,k].  grid 1024 x 256 (8192 waves = tiles).
__global__ void hw_kernel(const float* __restrict__ Hm,
                          const float* __restrict__ wl,
                          float* __restrict__ HW) {
  __shared__ float sW[HH * HH];
  const int tid = threadIdx.x;
  for (int i = tid; i < HH * HH; i += 256) sW[i] = wl[i];
  __syncthreads();
  const int wave = tid >> 5, lane = tid & 31;
  const int tile = blockIdx.x * 8 + wave;
  const int rt = tile >> 1, ctile = tile & 1;
  const int half = lane >> 4, c = lane & 15;
  const float* arow = Hm + (size_t)(rt * 16 + c) * HH + 2 * half;
  v8f acc = {0, 0, 0, 0, 0, 0, 0, 0};
#pragma unroll
  for (int kc = 0; kc < HH / 4; ++kc) {
    const int k0 = kc * 4;
    v2f a = *(const v2f*)(arow + k0);
    v2f b;
    b[0] = sW[(k0 + 2 * half) * HH + ctile * 16 + c];
    b[1] = sW[(k0 + 2 * half + 1) * HH + ctile * 16 + c];
    acc = wmma4(a, b, acc);
  }
#pragma unroll
  for (int rr = 0; rr < 8; ++rr)
    HW[(size_t)(rt * 16 + rr + 8 * half) * HH + ctile * 16 + c] = acc[rr];
}

// T1[16 x 2048] = Uo^T @ HW'  where HW'[n, b*32+h] = HW[(b*NN+n)*32+h].
// grid 16 x 256 (128 waves = 128 column tiles).
__global__ void t1_kernel(const float* __restrict__ Uo,
                          const float* __restrict__ HW,
                          float* __restrict__ T1) {
  const int tid = threadIdx.x, wave = tid >> 5, lane = tid & 31;
  const int ct = blockIdx.x * 8 + wave;
  const int half = lane >> 4, c = lane & 15;
  const int cg = ct * 16 + c;
  const int bb = cg >> 5, h2 = cg & 31;
  const float* hwb = HW + (size_t)bb * (NN * HH) + h2;
  v8f acc = {0, 0, 0, 0, 0, 0, 0, 0};
  for (int kc = 0; kc < NN / 4; ++kc) {
    const int kk = kc * 4 + 2 * half;
    v2f a; a[0] = Uo[kk * NEIG + c];       a[1] = Uo[(kk + 1) * NEIG + c];
    v2f b; b[0] = hwb[(size_t)kk * HH];    b[1] = hwb[(size_t)(kk + 1) * HH];
    acc = wmma4(a, b, acc);
  }
#pragma unroll
  for (int rr = 0; rr < 8; ++rr)
    T1[(rr + 8 * half) * (BSZ * HH) + cg] = acc[rr];
}

// H[b,n,h] += relu( (Us @ T1)[n, b*32+h] ), fused store. grid 1024 x 256.
__global__ void t2_kernel(const float* __restrict__ Us,
                          const float* __restrict__ T1,
                          float* __restrict__ Hm) {
  const int tid = threadIdx.x, wave = tid >> 5, lane = tid & 31;
  const int tile = blockIdx.x * 8 + wave;     // 0..8191
  const int nt = tile >> 7, ct = tile & 127;
  const int half = lane >> 4, c = lane & 15;
  const int cg = ct * 16 + c;
  const float* arow = Us + (size_t)(nt * 16 + c) * NEIG + 2 * half;
  v8f acc = {0, 0, 0, 0, 0, 0, 0, 0};
#pragma unroll
  for (int kc = 0; kc < NEIG / 4; ++kc) {
    const int k0 = kc * 4;
    v2f a = *(const v2f*)(arow + k0);
    const int kk = k0 + 2 * half;
    v2f b; b[0] = T1[kk * (BSZ * HH) + cg]; b[1] = T1[(kk + 1) * (BSZ * HH) + cg];
    acc = wmma4(a, b, acc);
  }
  const int bb = cg >> 5, h2 = cg & 31;
#pragma unroll
  for (int rr = 0; rr < 8; ++rr) {
    const int n = nt * 16 + rr + 8 * half;
    const size_t idx = (size_t)bb * (NN * HH) + (size_t)n * HH + h2;
    float v = acc[rr];
    Hm[idx] = (v > 0.0f ? v : 0.0f) + Hm[idx];
  }
}

// ===========================================================================
// 5) Output head: out[b,n,o] = H[b,n,:] @ lin_w[:,o] + lin_b[o]
// ===========================================================================
__global__ void out_kernel(const float* __restrict__ Hm,
                           const float* __restrict__ lw,
                           const float* __restrict__ lb,
                           float* __restrict__ out) {
  const int idx = blockIdx.x * 256 + threadIdx.x;   // BN*12 exact
  const int bn = idx / 12, o = idx % 12;
  const float* hp = Hm + (size_t)bn * HH;
  float s = lb[o];
#pragma unroll
  for (int h = 0; h < HH; ++h) s += hp[h] * lw[h * 12 + o];
  out[idx] = s;
}

// ===========================================================================
// Host-side orchestration
// ===========================================================================
extern "C" void kernel_launch(void* const* d_in, const int* in_sizes, int n_in,
                              void* d_out, int out_size, void* d_ws, size_t ws_size,
                              hipStream_t stream) {
  (void)n_in; (void)out_size; (void)ws_size;
  const float* x      = (const float*)d_in[0];
  const int*   ei     = (const int*)  d_in[1];
  const float* ew     = (const float*)d_in[2];
  const float* gwih   = (const float*)d_in[3];
  const float* gwhh   = (const float*)d_in[4];
  const float* gbih   = (const float*)d_in[5];
  const float* gbhh   = (const float*)d_in[6];
  const float* convw  = (const float*)d_in[7];
  const float* linw   = (const float*)d_in[8];
  const float* linb   = (const float*)d_in[9];
  float* out = (float*)d_out;
  const int E = in_sizes[2];   // E_RAND + N = 33792

  // ---- workspace layout (floats) ----
  float* ws   = (float*)d_ws;
  float* LAP  = ws;                         // 1M
  float* LAPT = LAP  + (size_t)NN * NN;     // 1M
  float* Hbuf = LAPT + (size_t)NN * NN;     // 2M  (B*N*H)
  float* HWb  = Hbuf + (size_t)BN * HH;     // 2M  (also hosts A during build)
  float* Abuf = HWb;                        // 1M subset of HWb region
  float* deg  = HWb  + (size_t)BN * HH;     // 1024
  float* XA   = deg  + NN;                  // 16K
  float* XB   = XA   + NN * NEIG;           // 16K
  float* Yb   = XB   + NN * NEIG;           // 16K
  float* T16  = Yb   + NN * NEIG;           // 256
  float* Vs   = T16  + NEIG * NEIG;         // 256
  float* lam  = Vs   + NEIG * NEIG;         // 16
  float* Uo   = lam  + NEIG;                // 16K
  float* Us   = Uo   + NN * NEIG;           // 16K
  float* T1   = Us   + NN * NEIG;           // 32K (16 x 2048)

  // ---- 1) GRU ----
  gru_kernel<<<BN / 128, 128, 0, stream>>>(x, gwih, gwhh, gbih, gbhh, Hbuf);

  // ---- 2) Laplacian (A region overlays HWb; HWb unused until conv phase) ----
  zero_kernel<<<(NN * NN) / 256, 256, 0, stream>>>(Abuf, NN * NN);
  scatter_kernel<<<(E + 255) / 256, 256, 0, stream>>>(ei, ew, Abuf, E);
  degree_kernel<<<NN / 256, 256, 0, stream>>>(Abuf, deg);
  lap_kernel<<<(NN * NN) / 256, 256, 0, stream>>>(Abuf, deg, LAP, LAPT);

  // ---- 3) top-16 subspace of G = LAP*LAP^T via block power iteration ----
  initx_kernel<<<(NN * NEIG) / 256, 256, 0, stream>>>(XA);
  qr_kernel<<<1, 256, 0, stream>>>(XA);
  float* cur = XA; float* nxt = XB;
  const int NITER = 48;
  for (int it = 0; it < NITER; ++it) {
    gemm_nk16_kernel<<<16, 128, 0, stream>>>(LAPT, cur, Yb);  // Y = L^T X
    gemm_nk16_kernel<<<16, 128, 0, stream>>>(LAP, Yb, nxt);   // Z = L Y = G X
    qr_kernel<<<1, 256, 0, stream>>>(nxt);
    float* t = cur; cur = nxt; nxt = t;
  }
  // Rayleigh-Ritz: T16 = X^T G X, eigendecompose, rotate basis.
  gemm_nk16_kernel<<<16, 128, 0, stream>>>(LAPT, cur, Yb);
  gemm_nk16_kernel<<<16, 128, 0, stream>>>(LAP, Yb, nxt);     // nxt = G X
  t16_kernel<<<1, 256, 0, stream>>>(cur, nxt, T16);
  jacobi16_kernel<<<1, 32, 0, stream>>>(T16, Vs, lam);
  ritz_kernel<<<(NN * NEIG) / 256, 256, 0, stream>>>(cur, Vs, lam, Uo, Us);

  // ---- 4) three factored spectral conv layers ----
  for (int l = 0; l < 3; ++l) {
    hw_kernel<<<1024, 256, 0, stream>>>(Hbuf, convw + (size_t)l * HH * HH, HWb);
    t1_kernel<<<16, 256, 0, stream>>>(Uo, HWb, T1);
    t2_kernel<<<1024, 256, 0, stream>>>(Us, T1, Hbuf);
  }

  // ---- 5) output head ----
  out_kernel<<<(BN * 12) / 256, 256, 0, stream>>>(Hbuf, linw, linb, out);
}